// PolicyNetwork_53334903881955
// MI455X (gfx1250) — compile-verified
//
#include <hip/hip_runtime.h>
#include <stdint.h>

// ---------------------------------------------------------------------------
// GAT policy network for MI455X (gfx1250, wave32).
//  - Dense GEMMs (z = h@W, action MLP) use V_WMMA_F32_16X16X4_F32: exact f32,
//    and the GEMMs are <1% of runtime (edge scatter is L2-bound), so no need
//    to trade precision for f16 WMMA throughput.
//  - Edge softmax/aggregation: monotonic-key atomicMax for segment-max,
//    atomicAdd for segment-sum, wave-per-edge scatter of alpha*z[src].
//    Working set (z 5MB, agg 5MB, edges 10MB) lives in the 192MB L2.
// ---------------------------------------------------------------------------

#define NNODES 10000
#define NEDGES 640000
#define E2     (NEDGES + NNODES)   // edges + self loops
#define HID    128
#define NLAYERS 3
#define NACT   4096
#define NEG_SLOPE 0.2f

typedef float v2f __attribute__((ext_vector_type(2)));
typedef float v8f __attribute__((ext_vector_type(8)));

static __device__ __forceinline__ v8f wmma_f32(v2f a, v2f b, v8f c) {
  // D = A(16x4) * B(4x16) + C(16x16), all f32. 8-arg VOP3P form:
  // (neg_a, A, neg_b, B, c_mod, C, reuse_a, reuse_b)
  return __builtin_amdgcn_wmma_f32_16x16x4_f32(
      false, a, false, b, (short)0, c, false, false);
}

// order-preserving f32 <-> u32 map for atomic float max (key 0 == "empty")
static __device__ __forceinline__ unsigned f32_key(float x) {
  unsigned b = __float_as_uint(x);
  return (b & 0x80000000u) ? ~b : (b | 0x80000000u);
}
static __device__ __forceinline__ float f32_unkey(unsigned u) {
  return (u & 0x80000000u) ? __uint_as_float(u & 0x7FFFFFFFu)
                           : __uint_as_float(~u);
}

// ---------------------------------------------------------------------------
// z[16 x 128] tile = h_tile[16 x 128] @ W[128 x 128], one block per M-tile.
// 4 waves, 2 N-tiles of 16 per wave. A staged in LDS (pad 132: 4-dword stride
// so the two lane-halves hit disjoint banks).
// ---------------------------------------------------------------------------
__global__ __launch_bounds__(128)
void gemm_z_kernel(const float* __restrict__ h, const float* __restrict__ W,
                   float* __restrict__ z) {
  __shared__ float As[16][132];
  const int tid = threadIdx.x;
  const int m0  = blockIdx.x * 16;
#pragma unroll
  for (int i = 0; i < 16; ++i) {
    int idx = tid + 128 * i;
    int r = idx >> 7, c = idx & 127;
    As[r][c] = h[(m0 + r) * HID + c];
  }
  __syncthreads();
  const int wave = tid >> 5, lane = tid & 31;
  const int half = lane >> 4, ln = lane & 15;
#pragma unroll
  for (int t = 0; t < 2; ++t) {
    const int n0 = (wave * 2 + t) * 16;
    const int n  = n0 + ln;
    v8f acc = {};
#pragma unroll 4
    for (int k = 0; k < HID; k += 4) {
      v2f a, b;
      a.x = As[ln][k + 2 * half];          // A: lanes 0-15 K,K+1; 16-31 K+2,K+3
      a.y = As[ln][k + 2 * half + 1];
      b.x = W[(k + 2 * half) * HID + n];   // B: same K split, N = lane%16
      b.y = W[(k + 2 * half + 1) * HID + n];
      acc = wmma_f32(a, b, acc);
    }
#pragma unroll
    for (int g = 0; g < 8; ++g)            // D: M = g + 8*half, N = lane%16
      z[(m0 + g + 8 * half) * HID + n] = acc[g];
  }
}

// one wave per node: es = z_row . a_src, ed = z_row . a_dst
__global__ __launch_bounds__(256)
void node_dots_kernel(const float* __restrict__ z, const float* __restrict__ as,
                      const float* __restrict__ ad, float* __restrict__ es,
                      float* __restrict__ ed) {
  int node = blockIdx.x * (blockDim.x >> 5) + (threadIdx.x >> 5);
  if (node >= NNODES) return;
  int lane = threadIdx.x & 31;
  float s1 = 0.f, s2 = 0.f;
#pragma unroll
  for (int i = 0; i < 4; ++i) {
    int f = lane + 32 * i;
    float zv = z[node * HID + f];
    s1 += zv * as[f];
    s2 += zv * ad[f];
  }
#pragma unroll
  for (int o = 16; o > 0; o >>= 1) {
    s1 += __shfl_xor(s1, o);
    s2 += __shfl_xor(s2, o);
  }
  if (lane == 0) { es[node] = s1; ed[node] = s2; }
}

__global__ __launch_bounds__(256)
void edge_logit_max_kernel(const long long* __restrict__ ei,
                           const float* __restrict__ es,
                           const float* __restrict__ ed,
                           float* __restrict__ ev, unsigned* __restrict__ mkey) {
  int e = blockIdx.x * blockDim.x + threadIdx.x;
  if (e >= E2) return;
  int s, d;
  if (e < NEDGES) { s = (int)ei[e]; d = (int)ei[NEDGES + e]; }
  else            { s = d = e - NEDGES; }
  float v = es[s] + ed[d];
  v = v > 0.f ? v : NEG_SLOPE * v;       // leaky_relu
  ev[e] = v;
  atomicMax(&mkey[d], f32_key(v));
}

__global__ __launch_bounds__(256)
void finalize_max_kernel(const unsigned* __restrict__ mkey,
                         float* __restrict__ m) {
  int n = blockIdx.x * blockDim.x + threadIdx.x;
  if (n >= NNODES) return;
  unsigned u = mkey[n];
  m[n] = (u == 0u) ? 0.f : f32_unkey(u); // matches isfinite() guard in ref
}

__global__ __launch_bounds__(256)
void edge_exp_sum_kernel(const long long* __restrict__ ei,
                         float* __restrict__ ev, const float* __restrict__ m,
                         float* __restrict__ denom) {
  int e = blockIdx.x * blockDim.x + threadIdx.x;
  if (e >= E2) return;
  int d = (e < NEDGES) ? (int)ei[NEDGES + e] : e - NEDGES;
  float ex = expf(ev[e] - m[d]);
  ev[e] = ex;                            // in-place: logits -> exp
  atomicAdd(&denom[d], ex);
}

// one wave per edge: agg[dst] += (ex/denom[dst]) * z[src], 4 feats per lane
__global__ __launch_bounds__(256)
void edge_aggregate_kernel(const long long* __restrict__ ei,
                           const float* __restrict__ ex,
                           const float* __restrict__ denom,
                           const float* __restrict__ z,
                           float* __restrict__ agg) {
  int e = blockIdx.x * (blockDim.x >> 5) + (threadIdx.x >> 5);
  if (e >= E2) return;
  int lane = threadIdx.x & 31;
  int s, d;
  if (e < NEDGES) { s = (int)ei[e]; d = (int)ei[NEDGES + e]; }
  else            { s = d = e - NEDGES; }
  float alpha = ex[e] / denom[d];
  const float* zs = z + (size_t)s * HID;
  float* od = agg + (size_t)d * HID;
#pragma unroll
  for (int i = 0; i < 4; ++i) {
    int f = lane + 32 * i;
    atomicAdd(&od[f], alpha * zs[f]);
  }
}

__global__ __launch_bounds__(256)
void bias_relu_kernel(const float* __restrict__ agg,
                      const float* __restrict__ b, float* __restrict__ hout) {
  int i = blockIdx.x * blockDim.x + threadIdx.x;
  if (i >= NNODES * HID) return;
  float v = agg[i] + b[i & (HID - 1)];
  hout[i] = v > 0.f ? v : 0.f;
}

// ---------------------------------------------------------------------------
// Action MLP: block of 4 waves handles 16 actions. Gather h[src], h[dst] into
// LDS, hidden = relu([hs|hd|type]@W1 + b1) via two K=128 WMMA chains, then
// score = hidden.W2 + b2 reduced through LDS.
// ---------------------------------------------------------------------------
__global__ __launch_bounds__(128)
void action_mlp_kernel(const float* __restrict__ h,
                       const long long* __restrict__ asrc,
                       const long long* __restrict__ adst,
                       const int* __restrict__ atype,
                       const float* __restrict__ W1,
                       const float* __restrict__ b1,
                       const float* __restrict__ W2,
                       const float* __restrict__ b2,
                       float* __restrict__ out) {
  __shared__ float A0[16][132];
  __shared__ float A1[16][132];
  __shared__ float sScore[16];
  const int tid = threadIdx.x;
  const int a0  = blockIdx.x * 16;
#pragma unroll
  for (int i = 0; i < 16; ++i) {
    int idx = tid + 128 * i;
    int r = idx >> 7, c = idx & 127;
    A0[r][c] = h[(int)asrc[a0 + r] * HID + c];
    A1[r][c] = h[(int)adst[a0 + r] * HID + c];
  }
  if (tid < 16) sScore[tid] = 0.f;
  __syncthreads();
  const int wave = tid >> 5, lane = tid & 31;
  const int half = lane >> 4, ln = lane & 15;
#pragma unroll
  for (int t = 0; t < 2; ++t) {
    const int n0 = (wave * 2 + t) * 16;
    const int n  = n0 + ln;
    v8f acc = {};
#pragma unroll 4
    for (int k = 0; k < HID; k += 4) {       // h[src] @ W1[0:128]
      v2f a, b;
      a.x = A0[ln][k + 2 * half];
      a.y = A0[ln][k + 2 * half + 1];
      b.x = W1[(k + 2 * half) * HID + n];
      b.y = W1[(k + 2 * half + 1) * HID + n];
      acc = wmma_f32(a, b, acc);
    }
#pragma unroll 4
    for (int k = 0; k < HID; k += 4) {       // h[dst] @ W1[128:256]
      v2f a, b;
      a.x = A1[ln][k + 2 * half];
      a.y = A1[ln][k + 2 * half + 1];
      b.x = W1[(HID + k + 2 * half) * HID + n];
      b.y = W1[(HID + k + 2 * half + 1) * HID + n];
      acc = wmma_f32(a, b, acc);
    }
    const float w1t = W1[2 * HID * HID + n]; // type-flag row (row 256)
    const float bb  = b1[n];
    const float w2  = W2[n];
#pragma unroll
    for (int g = 0; g < 8; ++g) {
      int mrow = g + 8 * half;
      float v = acc[g] + (float)atype[a0 + mrow] * w1t + bb;
      v = v > 0.f ? v : 0.f;                 // relu(hidden)
      atomicAdd(&sScore[mrow], v * w2);      // hidden . W2 partial
    }
  }
  __syncthreads();
  if (tid < 16) out[a0 + tid] = sScore[tid] + b2[0];
}

// ---------------------------------------------------------------------------
extern "C" void kernel_launch(void* const* d_in, const int* in_sizes, int n_in,
                              void* d_out, int out_size, void* d_ws,
                              size_t ws_size, hipStream_t stream) {
  (void)in_sizes; (void)n_in; (void)out_size; (void)ws_size;
  const float*     x     = (const float*)d_in[0];
  const long long* ei    = (const long long*)d_in[1];   // int64 [2, E]
  const long long* asrc  = (const long long*)d_in[2];   // int64 [NACT]
  const long long* adst  = (const long long*)d_in[3];
  const int*       atype = (const int*)d_in[4];         // int32 [NACT]
  const float*     gat_W = (const float*)d_in[5];       // [3,128,128]
  const float*     gat_as= (const float*)d_in[6];       // [3,128]
  const float*     gat_ad= (const float*)d_in[7];
  const float*     gat_b = (const float*)d_in[8];
  const float*     W1    = (const float*)d_in[9];       // [257,128]
  const float*     b1    = (const float*)d_in[10];
  const float*     W2    = (const float*)d_in[11];      // [128,1]
  const float*     b2    = (const float*)d_in[12];
  float*           out   = (float*)d_out;

  // workspace carve-up (256B aligned), ~18 MB total
  char*  ws  = (char*)d_ws;
  size_t off = 0;
  auto carve = [&](size_t bytes) -> void* {
    void* p = ws + off;
    off += (bytes + 255) & ~(size_t)255;
    return p;
  };
  float*    hbuf  = (float*)carve((size_t)NNODES * HID * 4);
  float*    z     = (float*)carve((size_t)NNODES * HID * 4);
  float*    agg   = (float*)carve((size_t)NNODES * HID * 4);
  float*    es    = (float*)carve((size_t)NNODES * 4);
  float*    ed    = (float*)carve((size_t)NNODES * 4);
  unsigned* mkey  = (unsigned*)carve((size_t)NNODES * 4);
  float*    m     = (float*)carve((size_t)NNODES * 4);
  float*    denom = (float*)carve((size_t)NNODES * 4);
  float*    ev    = (float*)carve((size_t)E2 * 4);      // logits, then exp

  const float* hin = x;
  for (int l = 0; l < NLAYERS; ++l) {
    hipMemsetAsync(agg,   0, (size_t)NNODES * HID * 4, stream);
    hipMemsetAsync(mkey,  0, (size_t)NNODES * 4, stream);
    hipMemsetAsync(denom, 0, (size_t)NNODES * 4, stream);

    gemm_z_kernel<<<NNODES / 16, 128, 0, stream>>>(hin, gat_W + l * HID * HID, z);
    node_dots_kernel<<<(NNODES + 7) / 8, 256, 0, stream>>>(
        z, gat_as + l * HID, gat_ad + l * HID, es, ed);
    edge_logit_max_kernel<<<(E2 + 255) / 256, 256, 0, stream>>>(ei, es, ed, ev, mkey);
    finalize_max_kernel<<<(NNODES + 255) / 256, 256, 0, stream>>>(mkey, m);
    edge_exp_sum_kernel<<<(E2 + 255) / 256, 256, 0, stream>>>(ei, ev, m, denom);
    edge_aggregate_kernel<<<(E2 + 7) / 8, 256, 0, stream>>>(ei, ev, denom, z, agg);
    bias_relu_kernel<<<(NNODES * HID + 255) / 256, 256, 0, stream>>>(
        agg, gat_b + l * HID, hbuf);
    hin = hbuf;
  }

  action_mlp_kernel<<<NACT / 16, 128, 0, stream>>>(
      hbuf, asrc, adst, atype, W1, b1, W2, b2, out);
}